// LEXNet_42193758716503
// MI455X (gfx1250) — compile-verified
//
#include <hip/hip_runtime.h>
#include <hip/hip_bf16.h>

typedef float v2f __attribute__((ext_vector_type(2)));
typedef float v8f __attribute__((ext_vector_type(8)));
typedef int   v4i __attribute__((ext_vector_type(4)));
typedef int   v2i __attribute__((ext_vector_type(2)));
typedef __attribute__((address_space(1))) v4i gv4i;
typedef __attribute__((address_space(3))) v4i lv4i;
typedef __attribute__((address_space(1))) v2i gv2i;
typedef __attribute__((address_space(3))) v2i lv2i;

#define B_SZ   64
#define C_SZ   512
#define HW     196
#define P_SZ   2000
#define NC_SZ  200
#define EPSQ   1e-4f

#define BN     80          // prototypes per block (2000 = 25 * 80)
#define BK     16          // K chunk
#define NTP    5           // proto n-subtiles per wave (5*16 = 80)
#define MT     13          // hw m-subtiles = waves  (13*16 = 208 >= 196)
#define NPAD   208
#define TPB    (MT*32)     // 416 threads
#define FLT_BIG 3.402823466e+38f

// ---------------- p2[p] = sum_c proto[p][c]^2 ----------------
__global__ void p2_kernel(const float* __restrict__ proto, float* __restrict__ p2) {
  int wid  = (blockIdx.x * blockDim.x + threadIdx.x) >> 5;
  int lane = threadIdx.x & 31;
  if (wid >= P_SZ) return;
  const float* row = proto + (size_t)wid * C_SZ;
  float s = 0.f;
  for (int c = lane; c < C_SZ; c += 32) { float v = row[c]; s += v * v; }
  for (int off = 16; off; off >>= 1) s += __shfl_xor(s, off, 32);
  if (lane == 0) p2[wid] = s;
}

// ---------------- x2[b][hw] = sum_c x[b][c][hw]^2 ----------------
__global__ void x2_kernel(const float* __restrict__ x, float* __restrict__ x2) {
  int b = blockIdx.x;
  int t = threadIdx.x;
  if (t >= HW) return;
  const float* xb = x + (size_t)b * C_SZ * HW + t;
  float s = 0.f;
  for (int c = 0; c < C_SZ; ++c) { float v = xb[(size_t)c * HW]; s += v * v; }
  x2[b * HW + t] = s;
}

// ---- async global->LDS staging of one K-chunk ----
// x tile (A side): [kk][n] plain rows, hw-contiguous  -> b128 copies
// proto tile (B side): pair-interleaved [(k/2)][p][k&1] -> b64 copies (k-contiguous in global)
__device__ __forceinline__ void issue_chunk_async(const float* __restrict__ xb,
                                                  const float* __restrict__ pb,
                                                  int k0, float* sXbuf, float* sPbuf,
                                                  int tid) {
  for (int idx = tid; idx < BK * 49; idx += TPB) {           // 16 rows x 49 b128
    int kk = idx / 49;
    int n4 = (idx - kk * 49) * 4;
    __builtin_amdgcn_global_load_async_to_lds_b128(
        (gv4i*)(xb + (size_t)(k0 + kk) * HW + n4),
        (lv4i*)(sXbuf + kk * NPAD + n4), 0, 0);
  }
  for (int idx = tid; idx < BN * (BK / 2); idx += TPB) {     // 80 p x 8 k-pairs, b64
    int p = idx >> 3;
    int j = idx & 7;                                         // k-pair index
    __builtin_amdgcn_global_load_async_to_lds_b64(
        (gv2i*)(pb + (size_t)p * C_SZ + k0 + 2 * j),
        (lv2i*)(sPbuf + (j * BN + p) * 2), 0, 0);
  }
}

// ------- fused GEMM (xp) + distance + min over hw, via V_WMMA_F32_16X16X4_F32 -------
// D[m=hw][n=proto] ; wave w owns hw rows w*16..w*16+15 and all 80 protos (5 n-tiles)
__launch_bounds__(TPB)
__global__ void gemm_min_kernel(const float* __restrict__ x,
                                const float* __restrict__ proto,
                                const float* __restrict__ p2g,
                                const float* __restrict__ x2g,
                                float* __restrict__ md) {
  __shared__ float sX[2][BK * NPAD];        // A tile: [kk][hw], rows 196 valid (+12 garbage, masked)
  __shared__ float sP[2][(BK / 2) * BN * 2];// B tile: pair-interleaved protos
  __shared__ float sX2[HW];
  __shared__ float sP2[BN];
  __shared__ float sPart[MT * BN];          // per-wave partial mins

  const int b      = blockIdx.y;
  const int nblk   = blockIdx.x;
  const int tid    = threadIdx.x;
  const int wave   = tid >> 5;              // = hw m-tile
  const int lane   = tid & 31;
  const int lane16 = lane & 15;
  const int half   = lane >> 4;             // 0: K+0/1 & M rows v ; 1: K+2/3 & M rows v+8

  const float* xb = x + (size_t)b * C_SZ * HW;
  const float* pb = proto + (size_t)nblk * BN * C_SZ;

  issue_chunk_async(xb, pb, 0, sX[0], sP[0], tid);

  for (int i = tid; i < HW; i += TPB) sX2[i] = x2g[b * HW + i];
  for (int i = tid; i < BN; i += TPB) sP2[i] = p2g[nblk * BN + i];

  v8f acc[NTP];
  #pragma unroll
  for (int nt = 0; nt < NTP; ++nt)
    #pragma unroll
    for (int e = 0; e < 8; ++e) acc[nt][e] = 0.f;

  __builtin_amdgcn_s_wait_asynccnt(0);
  __syncthreads();

  int buf = 0;
  for (int k0 = 0; k0 < C_SZ; k0 += BK) {
    if (k0 + BK < C_SZ)
      issue_chunk_async(xb, pb, k0 + BK, sX[buf ^ 1], sP[buf ^ 1], tid);

    const float* Xb = sX[buf];
    const float* Pb = sP[buf];

    // A fragments (x): 4 per chunk; row = hw = wave*16+lane16, k-pair by lane half
    v2f af[4];
    #pragma unroll
    for (int j = 0; j < 4; ++j) {
      int bk = j * 4 + half * 2;
      af[j].x = Xb[bk * NPAD + wave * 16 + lane16];
      af[j].y = Xb[(bk + 1) * NPAD + wave * 16 + lane16];
    }

    #pragma unroll
    for (int j = 0; j < 4; ++j) {
      const int bk2 = j * 2 + half;         // (j*4 + half*2) / 2
      v2f bf[NTP];                          // single aligned b64 each, no repacking
      #pragma unroll
      for (int nt = 0; nt < NTP; ++nt)
        bf[nt] = *(const v2f*)(&Pb[(bk2 * BN + nt * 16 + lane16) * 2]);
      #pragma unroll
      for (int nt = 0; nt < NTP; ++nt)
        acc[nt] = __builtin_amdgcn_wmma_f32_16x16x4_f32(
            false, af[j], false, bf[nt], (short)0, acc[nt], false, false);
    }

    __builtin_amdgcn_s_wait_asynccnt(0);
    __syncthreads();
    buf ^= 1;
  }

  // epilogue: dist = relu(x2 - 2*xp + p2); min over hw (= M dim)
  float x2v[8]; bool mval[8];
  #pragma unroll
  for (int v = 0; v < 8; ++v) {
    int hw = wave * 16 + v + half * 8;
    mval[v] = (hw < HW);
    x2v[v]  = sX2[mval[v] ? hw : 0];
  }

  #pragma unroll
  for (int nt = 0; nt < NTP; ++nt) {
    float p2v = sP2[nt * 16 + lane16];
    float mn = FLT_BIG;
    #pragma unroll
    for (int v = 0; v < 8; ++v) {
      float d = fmaxf(x2v[v] - 2.f * acc[nt][v] + p2v, 0.f);
      if (mval[v]) mn = fminf(mn, d);
    }
    mn = fminf(mn, __shfl_xor(mn, 16, 32));   // combine lane halves (same proto, other 8 rows)
    if (lane < 16) sPart[wave * BN + nt * 16 + lane] = mn;
  }
  __syncthreads();

  for (int p = tid; p < BN; p += TPB) {
    float mn = sPart[p];
    #pragma unroll
    for (int w = 1; w < MT; ++w) mn = fminf(mn, sPart[w * BN + p]);
    md[(size_t)b * P_SZ + nblk * BN + p] = mn;
  }
}

// ---------------- logits[b][nc] = sum_p log((md+1)/(md+eps)) * w[nc][p] ----------------
__global__ void logits_kernel(const float* __restrict__ md,
                              const float* __restrict__ w,
                              float* __restrict__ logits) {
  __shared__ float act[P_SZ];
  int b = blockIdx.x;
  int tid = threadIdx.x;
  for (int p = tid; p < P_SZ; p += 256) {
    float m = md[(size_t)b * P_SZ + p];
    act[p] = logf((m + 1.0f) / (m + EPSQ));
  }
  __syncthreads();
  int wave = tid >> 5, lane = tid & 31;
  for (int nc = wave; nc < NC_SZ; nc += 8) {
    const float* wr = w + (size_t)nc * P_SZ;
    float s = 0.f;
    for (int p = lane; p < P_SZ; p += 32) s += act[p] * wr[p];
    for (int off = 16; off; off >>= 1) s += __shfl_xor(s, off, 32);
    if (lane == 0) logits[b * NC_SZ + nc] = s;
  }
}

extern "C" void kernel_launch(void* const* d_in, const int* in_sizes, int n_in,
                              void* d_out, int out_size, void* d_ws, size_t ws_size,
                              hipStream_t stream) {
  const float* x     = (const float*)d_in[0];
  const float* proto = (const float*)d_in[1];
  const float* lw    = (const float*)d_in[2];

  float* logits = (float*)d_out;                    // 64*200
  float* md     = (float*)d_out + B_SZ * NC_SZ;     // 64*2000
  float* p2     = (float*)d_ws;                     // 2000
  float* x2     = (float*)d_ws + P_SZ;              // 64*196

  p2_kernel<<<(P_SZ * 32 + 255) / 256, 256, 0, stream>>>(proto, p2);
  x2_kernel<<<B_SZ, 256, 0, stream>>>(x, x2);
  dim3 grid(P_SZ / BN, B_SZ);
  gemm_min_kernel<<<grid, TPB, 0, stream>>>(x, proto, p2, x2, md);
  logits_kernel<<<B_SZ, 256, 0, stream>>>(md, lw, logits);
}